// MultiHeadFAVORAttention_2241972929046
// MI455X (gfx1250) — compile-verified
//
#include <hip/hip_runtime.h>
#include <math.h>

// ---------------- constants (match reference) ----------------
#define BB 2
#define SS 4096
#define CC 1024
#define HH 16
#define DH 64
#define NRF 256
#define GG 3          // gamma scales
#define MSC 256       // features per scale
#define NGROUP 4
#define LNDM 64

// CDNA5 async global->LDS path (global_load_async_to_lds_b32 + s_wait_asynccnt)
#define USE_ASYNC_LDS 1

typedef __attribute__((ext_vector_type(2))) float v2f;
typedef __attribute__((ext_vector_type(8))) float v8f;

// ---------------- generic strided batched WMMA f32 GEMM ----------------
// C[z][m][n] = epi( alpha * sum_k A[z][m,k]*B[z][k,n] + bias[n] ) (+ Cold if accum)
// epi==1: v = cos(alpha*acc + bias[n]) * cscale
static constexpr int BM = 128, BN = 64, BK = 16;
static constexpr int LDAS = BK + 1;   // LDS pad (odd stride -> conflict free frag reads)
static constexpr int LDBS = BN + 4;

// one 4-byte element: global -> LDS (async on CDNA5, staged fallback otherwise)
__device__ __forceinline__ void tile_ld(float* ldst, const float* g)
{
#if USE_ASYNC_LDS
    unsigned lo = (unsigned)(unsigned long long)ldst;   // low 32b of LDS generic addr
    asm volatile("global_load_async_to_lds_b32 %0, %1, off"
                 :: "v"(lo), "v"(g) : "memory");
#else
    *ldst = *g;
#endif
}

__device__ __forceinline__ void wait_async0()
{
#if USE_ASYNC_LDS
    asm volatile("s_wait_asynccnt 0x0" ::: "memory");
#endif
}

// 4 chained WMMAs per fragment pair over one BK=16 LDS tile
__device__ __forceinline__ void compute_tiles(const float* __restrict__ AsT,
                                              const float* __restrict__ BsT,
                                              v8f acc[2][2],
                                              int wm, int wn, int lr, int kq)
{
    #pragma unroll
    for (int kk = 0; kk < BK; kk += 4) {
        v2f a[2], b[2];
        #pragma unroll
        for (int s = 0; s < 2; ++s) {
            int mrow = wm * 32 + s * 16 + lr;
            a[s].x = AsT[mrow * LDAS + kk + kq];
            a[s].y = AsT[mrow * LDAS + kk + kq + 1];
            int ncol = wn * 32 + s * 16 + lr;
            b[s].x = BsT[(kk + kq)     * LDBS + ncol];
            b[s].y = BsT[(kk + kq + 1) * LDBS + ncol];
        }
        #pragma unroll
        for (int i = 0; i < 2; ++i)
            #pragma unroll
            for (int j = 0; j < 2; ++j)
                acc[i][j] = __builtin_amdgcn_wmma_f32_16x16x4_f32(
                    false, a[i], false, b[j], (short)0, acc[i][j], false, false);
    }
}

__global__ __launch_bounds__(256)
void gemm_wmma_f32(const float* __restrict__ A, const float* __restrict__ B,
                   float* __restrict__ C,
                   int M, int N, int K,
                   long long sAm, long long sAk,
                   long long sBk, long long sBn,
                   long long ldc,
                   long long bsA, long long bsB, long long bsC,
                   const float* __restrict__ bias,
                   float alpha, float cscale, int epi, int accum)
{
    __shared__ float As[2][BM * LDAS];
    __shared__ float Bs[2][BK * LDBS];

    const int tid  = threadIdx.x;
    const int lane = tid & 31;
    const int wv   = tid >> 5;      // 0..7
    const int wm   = wv & 3;        // wave M tile (0..3) -> 32 rows each
    const int wn   = wv >> 2;       // wave N tile (0..1) -> 32 cols each

    const long long z = blockIdx.z;
    A += z * bsA;  B += z * bsB;  C += z * bsC;

    const int m0 = blockIdx.y * BM;
    const int n0 = blockIdx.x * BN;

    v8f acc[2][2] = {};

    // lane split per ISA 16x4 layout: lanes 0-15 hold K=0,1 ; lanes 16-31 hold K=2,3
    const int hi = lane >> 4;
    const int lr = lane & 15;
    const int kq = hi * 2;

    const bool full = (m0 + BM <= M) && (n0 + BN <= N) && ((K & (BK - 1)) == 0);

    if (full) {
        // ---- async double-buffered pipeline ----
        // issue tile for k0 into buffer p (no bounds checks: full tile)
        auto issue = [&](int k0, int p) {
            #pragma unroll
            for (int i = 0; i < 8; ++i) {           // A: 128x16, 8 per thread
                int idx = tid + i * 256;
                int am = idx >> 4, ak = idx & 15;
                tile_ld(&As[p][am * LDAS + ak],
                        A + (long long)(m0 + am) * sAm + (long long)(k0 + ak) * sAk);
            }
            #pragma unroll
            for (int i = 0; i < 4; ++i) {           // B: 16x64, 4 per thread
                int idx = tid + i * 256;
                int bk = idx >> 6, bn = idx & 63;
                tile_ld(&Bs[p][bk * LDBS + bn],
                        B + (long long)(k0 + bk) * sBk + (long long)(n0 + bn) * sBn);
            }
            // warm GL2 two tiles ahead (global_prefetch_b8)
            int kpf = k0 + 2 * BK;
            if (kpf < K) {
                __builtin_prefetch(A + (long long)(m0 + (tid >> 4)) * sAm +
                                       (long long)(kpf + (tid & 15)) * sAk, 0, 3);
            }
        };

        issue(0, 0);
        wait_async0();
        __syncthreads();
        int p = 0;
        for (int k0 = 0; k0 < K; k0 += BK) {
            if (k0 + BK < K) issue(k0 + BK, p ^ 1);     // overlap fetch with math
            compute_tiles(As[p], Bs[p], acc, wm, wn, lr, kq);
            wait_async0();                               // next buffer landed
            __syncthreads();                             // visible to all waves
            p ^= 1;
        }
    } else {
        // ---- staged fallback with bounds checks ----
        for (int k0 = 0; k0 < K; k0 += BK) {
            __syncthreads();
            #pragma unroll
            for (int i = 0; i < 8; ++i) {
                int idx = tid + i * 256;
                int am = idx >> 4, ak = idx & 15;
                int gm = m0 + am, gk = k0 + ak;
                float v = 0.f;
                if (gm < M && gk < K) v = A[(long long)gm * sAm + (long long)gk * sAk];
                As[0][am * LDAS + ak] = v;
            }
            #pragma unroll
            for (int i = 0; i < 4; ++i) {
                int idx = tid + i * 256;
                int bk = idx >> 6, bn = idx & 63;
                int gk = k0 + bk, gn = n0 + bn;
                float v = 0.f;
                if (gk < K && gn < N) v = B[(long long)gk * sBk + (long long)gn * sBn];
                Bs[0][bk * LDBS + bn] = v;
            }
            __syncthreads();
            compute_tiles(As[0], Bs[0], acc, wm, wn, lr, kq);
        }
    }

    // C/D 16x16 layout: lanes 0-15 -> M=r, lanes 16-31 -> M=r+8 ; N = lane&15
    #pragma unroll
    for (int i = 0; i < 2; ++i) {
        #pragma unroll
        for (int j = 0; j < 2; ++j) {
            int gn = n0 + wn * 32 + j * 16 + lr;
            if (gn >= N) continue;
            float bv = bias ? bias[gn] : 0.f;
            #pragma unroll
            for (int r = 0; r < 8; ++r) {
                int gm = m0 + wm * 32 + i * 16 + (hi ? r + 8 : r);
                if (gm >= M) continue;
                float v = acc[i][j][r] * alpha + bv;
                if (epi == 1) v = __cosf(v) * cscale;
                long long off = (long long)gm * ldc + gn;
                if (accum) v += C[off];
                C[off] = v;
            }
        }
    }
}

// ---------------- elementwise / helper kernels ----------------

// zero-padded copy of x: [B, S+2, C]
__global__ void pad_x_kernel(const float* __restrict__ x, float* __restrict__ xp)
{
    long long i = (long long)blockIdx.x * blockDim.x + threadIdx.x;
    long long total = (long long)BB * (SS + 2) * CC;
    if (i >= total) return;
    int c = (int)(i % CC);
    long long t = i / CC;
    int sp = (int)(t % (SS + 2));
    int b  = (int)(t / (SS + 2));
    float v = 0.f;
    if (sp >= 1 && sp <= SS) v = x[((long long)b * SS + sp - 1) * CC + c];
    xp[i] = v;
}

// broadcast a length-C bias into [rows, C]
__global__ void bias_rows_kernel(float* __restrict__ out, const float* __restrict__ bias,
                                 long long rows)
{
    long long i = (long long)blockIdx.x * blockDim.x + threadIdx.x;
    if (i >= rows * CC) return;
    out[i] = bias[i % CC];
}

// [B,S,1024] -> [B,H,S,64] with optional rotate-half RoPE
__global__ void split_rope_kernel(const float* __restrict__ in, float* __restrict__ out,
                                  int applyRope)
{
    long long i = (long long)blockIdx.x * blockDim.x + threadIdx.x;
    long long total = (long long)BB * HH * SS * DH;
    if (i >= total) return;
    int d = (int)(i & (DH - 1));
    long long t = i >> 6;
    int s = (int)(t % SS);
    int h = (int)((t / SS) % HH);
    int b = (int)(t / ((long long)SS * HH));
    long long src = ((long long)b * SS + s) * CC + h * DH;
    float xv = in[src + d];
    if (applyRope) {
        int j = d & 31;
        float theta = (float)s * __expf(-(float)j * 0.28782313662425573f); // ln(1e4)/32
        float sn = __sinf(theta), cs = __cosf(theta);
        float other = in[src + ((d < 32) ? d + 32 : d - 32)];
        float rot = (d < 32) ? -other : other;
        xv = xv * cs + rot * sn;
    }
    out[i] = xv;
}

__global__ void lnorm_kernel(const float* __restrict__ lm, float* __restrict__ ln)
{
    int j = threadIdx.x;
    if (j >= LNDM) return;
    float s = 0.f;
    for (int d = 0; d < DH; ++d) { float v = lm[j * DH + d]; s += v * v; }
    ln[j] = s;
}

// in-place: dot -> exp(-(|k|^2 + |l_j|^2 - 2 dot)/64)
__global__ void nystrom_exp_kernel(const float* __restrict__ k, const float* __restrict__ ln,
                                   float* __restrict__ io, long long rows)
{
    long long r = (long long)blockIdx.x * blockDim.x + threadIdx.x;
    if (r >= rows) return;
    const float* kr = k + r * DH;
    float kk = 0.f;
    for (int d = 0; d < DH; ++d) kk += kr[d] * kr[d];
    float* o = io + r * LNDM;
    for (int j = 0; j < LNDM; ++j)
        o[j] = __expf(-(kk + ln[j] - 2.f * o[j]) * (1.f / (float)DH));
}

// in-place FAVOR+: dot -> exp(dot - |x|^2/2) / sqrt(NRF)
__global__ void favor_exp_kernel(const float* __restrict__ xr, float* __restrict__ io,
                                 long long rows)
{
    long long r = (long long)blockIdx.x * blockDim.x + threadIdx.x;
    if (r >= rows) return;
    const float* xv = xr + r * DH;
    float nrm = 0.f;
    for (int d = 0; d < DH; ++d) nrm += xv[d] * xv[d];
    nrm *= 0.5f;
    float* o = io + r * NRF;
    for (int m = 0; m < NRF; ++m)
        o[m] = __expf(o[m] - nrm) * 0.0625f;   // 1/sqrt(256)
}

// kpsum[bh][e] = sum_s kp[bh][s][e]   (block per bh, thread per e -> coalesced)
__global__ void colsum_kernel(const float* __restrict__ kp, float* __restrict__ out)
{
    int bh = blockIdx.x, e = threadIdx.x;
    const float* p = kp + (long long)bh * SS * NRF + e;
    float s = 0.f;
    for (int i = 0; i < SS; ++i) s += p[(long long)i * NRF];
    out[bh * NRF + e] = s;
}

// denom + divide + head-merge: attn[b,s,h*64+d] = num[bh,s,d] / (qp.kpsum + eps)
__global__ void combine_kernel(const float* __restrict__ qp, const float* __restrict__ kps,
                               const float* __restrict__ num, float* __restrict__ attn,
                               long long rows)
{
    long long r = (long long)blockIdx.x * blockDim.x + threadIdx.x;
    if (r >= rows) return;
    int s  = (int)(r % SS);
    int bh = (int)(r / SS);
    int h  = bh % HH, b = bh / HH;
    const float* q = qp + r * NRF;
    const float* ks = kps + (long long)bh * NRF;
    float den = 0.f;
    for (int e = 0; e < NRF; ++e) den += q[e] * ks[e];
    den += 1e-6f;
    const float* nm = num + r * DH;
    float* o = attn + ((long long)b * SS + s) * CC + h * DH;
    for (int d = 0; d < DH; ++d) o[d] = nm[d] / den;
}

// ---------------- host orchestration ----------------
static inline void launch_gemm(hipStream_t st,
    const float* A, const float* B, float* C,
    int M, int N, int K,
    long long sAm, long long sAk, long long sBk, long long sBn, long long ldc,
    long long bsA, long long bsB, long long bsC, int batches,
    const float* bias, float alpha, float cscale, int epi, int accum)
{
    dim3 grid((N + BN - 1) / BN, (M + BM - 1) / BM, batches);
    hipLaunchKernelGGL(gemm_wmma_f32, grid, dim3(256), 0, st,
                       A, B, C, M, N, K, sAm, sAk, sBk, sBn, ldc,
                       bsA, bsB, bsC, bias, alpha, cscale, epi, accum);
}

extern "C" void kernel_launch(void* const* d_in, const int* in_sizes, int n_in,
                              void* d_out, int out_size, void* d_ws, size_t ws_size,
                              hipStream_t stream)
{
    (void)in_sizes; (void)n_in; (void)out_size; (void)ws_size;
    const float* x      = (const float*)d_in[0];
    const float* conv_k = (const float*)d_in[1];
    const float* conv_b = (const float*)d_in[2];
    const float* rff_w  = (const float*)d_in[3];
    const float* rff_b  = (const float*)d_in[4];
    const float* proj_w = (const float*)d_in[5];
    const float* proj_b = (const float*)d_in[6];
    const float* omega  = (const float*)d_in[7];
    const float* lmarks = (const float*)d_in[8];
    const float* out_w  = (const float*)d_in[9];
    const float* out_b  = (const float*)d_in[10];
    float* outp = (float*)d_out;
    float* ws   = (float*)d_ws;

    const long long BS  = (long long)BB * SS;            // 8192
    const long long BHS = (long long)BB * HH * SS;       // 131072

    // workspace layout (floats), with reuse
    float* qkv   = ws;                                   // 3 * BHS*64
    float* hbuf  = qkv  + 3 * BHS * DH;                  // BS*1024 (conv out; later num)
    float* feats = hbuf + BS * CC;                       // BS*768
    float* k2    = feats + BS * (GG * MSC);              // BHS*64 (kdot->k2; later attn)
    float* qp    = k2   + BHS * DH;                      // BHS*256 (also dense scratch)
    float* kp    = qp   + BHS * NRF;                     // BHS*256 (also xpad scratch)
    float* kv    = kp   + BHS * NRF;                     // 32*256*64
    float* kps   = kv   + (long long)BB * HH * NRF * DH; // 32*256
    float* lnrm  = kps  + (long long)BB * HH * NRF;      // 64
    float* xpad  = kp;                                   // B*(S+2)*C fits in kp region
    float* dense = qp;                                   // BS*1024 fits in qp region
    float* numb  = hbuf;                                 // num reuses h scratch
    float* attn  = k2;                                   // attn reuses k2 region

    const float gammas[GG] = {0.5f, 1.0f, 2.0f};
    const float cscale = 0.08838834764831845f;           // sqrt(2/256)

    // padded input + landmark norms
    {
        long long tot = (long long)BB * (SS + 2) * CC;
        pad_x_kernel<<<dim3((unsigned)((tot + 255) / 256)), dim3(256), 0, stream>>>(x, xpad);
        lnorm_kernel<<<dim3(1), dim3(64), 0, stream>>>(lmarks, lnrm);
    }

    // ---- projections: grouped conv -> RFF cos -> dense -> split+RoPE ----
    for (int i = 0; i < 3; ++i) {
        // init h with conv bias, then 3 accumulating tap-GEMMs per group
        bias_rows_kernel<<<dim3((unsigned)((BS * CC + 255) / 256)), dim3(256), 0, stream>>>(
            hbuf, conv_b + i * CC, BS);
        for (int t = 0; t < 3; ++t)
            for (int g = 0; g < NGROUP; ++g)
                launch_gemm(stream,
                    xpad + (long long)t * CC + g * (CC / NGROUP),
                    conv_k + ((long long)(i * 3 + t) * (CC / NGROUP)) * CC + g * (CC / NGROUP),
                    hbuf + g * (CC / NGROUP),
                    SS, CC / NGROUP, CC / NGROUP,
                    CC, 1, CC, 1, CC,
                    (long long)(SS + 2) * CC, 0, (long long)SS * CC, BB,
                    nullptr, 1.f, 1.f, 0, 1);
        // multi-scale RFF: feats[:, g*256:(g+1)*256] = cos(h@rw * sqrt(2g) + rb) * cscale
        for (int gi = 0; gi < GG; ++gi)
            launch_gemm(stream,
                hbuf, rff_w + (long long)(i * GG + gi) * CC * MSC, feats + gi * MSC,
                (int)BS, MSC, CC,
                CC, 1, MSC, 1, GG * MSC,
                0, 0, 0, 1,
                rff_b + (long long)(i * GG + gi) * MSC,
                sqrtf(2.f * gammas[gi]), cscale, 1, 0);
        // dense: qkv_flat = feats @ proj_w + proj_b
        launch_gemm(stream,
            feats, proj_w + (long long)i * GG * MSC * CC, dense,
            (int)BS, CC, GG * MSC,
            GG * MSC, 1, CC, 1, CC,
            0, 0, 0, 1,
            proj_b + i * CC, 1.f, 1.f, 0, 0);
        // head split (+RoPE for q,k)
        split_rope_kernel<<<dim3((unsigned)((BHS * DH + 255) / 256)), dim3(256), 0, stream>>>(
            dense, qkv + (long long)i * BHS * DH, i < 2 ? 1 : 0);
    }

    const float* qh = qkv;
    const float* kh = qkv + BHS * DH;
    const float* vh = qkv + 2 * BHS * DH;
    const int NBATCH = BB * HH;

    // ---- Nystrom RBF features for k ----
    launch_gemm(stream, kh, lmarks, k2,
                SS, LNDM, DH,
                DH, 1, /*B = landmarks^T*/ 1, DH, LNDM,
                (long long)SS * DH, 0, (long long)SS * LNDM, NBATCH,
                nullptr, 1.f, 1.f, 0, 0);
    nystrom_exp_kernel<<<dim3((unsigned)((BHS + 255) / 256)), dim3(256), 0, stream>>>(
        kh, lnrm, k2, BHS);

    // ---- FAVOR+ features ----
    launch_gemm(stream, qh, omega, qp,
                SS, NRF, DH, DH, 1, NRF, 1, NRF,
                (long long)SS * DH, 0, (long long)SS * NRF, NBATCH,
                nullptr, 1.f, 1.f, 0, 0);
    favor_exp_kernel<<<dim3((unsigned)((BHS + 255) / 256)), dim3(256), 0, stream>>>(qh, qp, BHS);

    launch_gemm(stream, k2, omega, kp,
                SS, NRF, DH, DH, 1, NRF, 1, NRF,
                (long long)SS * DH, 0, (long long)SS * NRF, NBATCH,
                nullptr, 1.f, 1.f, 0, 0);
    favor_exp_kernel<<<dim3((unsigned)((BHS + 255) / 256)), dim3(256), 0, stream>>>(k2, kp, BHS);

    // ---- linear attention: kpsum, kv = kp^T @ v, num = qp @ kv ----
    colsum_kernel<<<dim3(NBATCH), dim3(NRF), 0, stream>>>(kp, kps);

    launch_gemm(stream, kp, vh, kv,
                NRF, DH, SS,
                /*A = kp^T*/ 1, NRF, DH, 1, DH,
                (long long)SS * NRF, (long long)SS * DH, (long long)NRF * DH, NBATCH,
                nullptr, 1.f, 1.f, 0, 0);

    launch_gemm(stream, qp, kv, numb,
                SS, DH, NRF,
                NRF, 1, DH, 1, DH,
                (long long)SS * NRF, (long long)NRF * DH, (long long)SS * DH, NBATCH,
                nullptr, 1.f, 1.f, 0, 0);

    combine_kernel<<<dim3((unsigned)((BHS + 255) / 256)), dim3(256), 0, stream>>>(
        qp, kps, numb, attn, BHS);

    // ---- output projection ----
    launch_gemm(stream, attn, out_w, outp,
                (int)BS, CC, CC,
                CC, 1, CC, 1, CC,
                0, 0, 0, 1,
                out_b, 1.f, 1.f, 0, 0);
}